// RadiusGraph_67997922230591
// MI455X (gfx1250) — compile-verified
//
#include <hip/hip_runtime.h>

// ---------------------------------------------------------------------------
// RadiusGraph on MI455X (gfx1250, wave32).
// Distance tiles computed with V_WMMA_F32_16X16X4_F32:
//   D[m][n] = refvec[m] . (-2*qvec[n]) + |refvec[m]|^2          (C preload)
// where vec = (x, y, z, S*batch), SCALE_B=4 so cross-batch pairs get +>=16
// on d^2 (radius^2 = 0.25): the batch mask folds into the distance.
// Hot path per element is a single v_cmpx against tmin=min(thr, bestd[15]).
// The N-tile loop is an explicit two-body software pipeline: buffers A0/A1
// alternate by name (no register rotation) and each tile's loads are issued
// one full tile-process before their use, so waits are partial.
// ---------------------------------------------------------------------------

typedef __attribute__((ext_vector_type(2))) float v2f;
typedef __attribute__((ext_vector_type(8))) float v8f;

#define KNN      16
#define SCALE_B  4.0f
#define SENT     3.402823466e+38f

// ---- Kernel 1: pack ref rows into WMMA A-operand layout + |r|^2 -----------
// A layout (32-bit 16x4): lane L holds A[M=L%16][K=2*(L>=16)+v] in VGPR v.
// packedA[t*64 + 2*L + v] = that element, so the main loop is one b64 load.
__global__ void rg_pack(const float4* __restrict__ ref,
                        float* __restrict__ packedA,
                        float* __restrict__ r2, int N) {
  int j = blockIdx.x * blockDim.x + threadIdx.x;
  if (j >= N) return;
  float4 r = ref[j];                 // (batch, x, y, z)
  float bb = r.x * SCALE_B;
  float x = r.y, y = r.z, z = r.w;
  r2[j] = x * x + y * y + z * z + bb * bb;
  int t = j >> 4, m = j & 15;
  float* p = packedA + (size_t)t * 64;
  p[2 * m + 0]        = x;           // lane m   (half 0): K0, K1
  p[2 * m + 1]        = y;
  p[2 * (m + 16) + 0] = z;           // lane m+16 (half 1): K2, K3
  p[2 * (m + 16) + 1] = bb;
}

// ---- Kernel 2: main WMMA loop. One wave = one 16-query tile x one N-slice -
__global__ __launch_bounds__(256) void rg_main(
    const float4* __restrict__ query, const float* __restrict__ radius,
    const float* __restrict__ packedA, const float* __restrict__ r2,
    float* __restrict__ part_d, int* __restrict__ part_i, int N, int S) {
  __shared__ float sd[8][32][KNN];
  __shared__ int   si[8][32][KNN];

  const int lane  = threadIdx.x & 31;
  const int wib   = threadIdx.x >> 5;                    // wave in block
  const int wid   = blockIdx.x * (blockDim.x >> 5) + wib;
  const int qt    = wid / S;                             // query tile
  const int slice = wid % S;                             // N slice
  const int n     = lane & 15;                           // query column
  const int h     = lane >> 4;                           // lane half
  const int q     = qt * 16 + n;

  // B operand (4x16, constant all loop): lane holds B[K=2h+v][N=n] = -2*qvec.
  float4 qr = query[q];
  float qb = qr.x * SCALE_B;
  float qx = qr.y, qy = qr.z, qz = qr.w;
  float rad = radius[q];
  float thr = rad * rad - (qx * qx + qy * qy + qz * qz + qb * qb);
  v2f b;
  b.x = h ? (-2.0f * qz) : (-2.0f * qx);
  b.y = h ? (-2.0f * qb) : (-2.0f * qy);

  float bestd[KNN];
  int   besti[KNN];
#pragma unroll
  for (int i = 0; i < KNN; ++i) { bestd[i] = SENT; besti[i] = -1; }

  const int tilesN = N >> 4;
  const int tps    = tilesN / S;                         // even by design
  const int t0     = slice * tps;
  const int t1     = t0 + tps;

  auto loadT = [&](int t, v2f& a, float4& c0, float4& c1) {
    const float* pA = packedA + (size_t)t * 64 + 2 * lane;
    a  = *(const v2f*)pA;
    const float* pC = r2 + t * 16 + 8 * h;
    c0 = *(const float4*)pC;
    c1 = *(const float4*)(pC + 4);
  };

  auto process = [&](int t, v2f a, float4 c0, float4 c1) {
    v8f c = {c0.x, c0.y, c0.z, c0.w, c1.x, c1.y, c1.z, c1.w};
    v8f d = __builtin_amdgcn_wmma_f32_16x16x4_f32(
        false, a, false, b, (short)0, c, false, false);
    // One compare per element. Stale tmin within a tile is safe: the bubble
    // insert is a no-op for keys that don't beat the current list.
    const float tmin = fminf(thr, bestd[KNN - 1]);
    const int ibase = (t << 4) + 8 * h;
#pragma unroll
    for (int v = 0; v < 8; ++v) {
      float key = d[v];
      if (key <= tmin) {                          // rare: s_cbranch_execz path
        int id = ibase + v;
#pragma unroll
        for (int i = 0; i < KNN; ++i) {           // register bubble-insert
          float od = bestd[i]; int oi = besti[i];
          bool sw = key < od;
          bestd[i] = sw ? key : od;  besti[i] = sw ? id : oi;
          key      = sw ? od : key;  id       = sw ? oi : id;
        }
      }
    }
  };

  // Two-body software pipeline: named buffer pairs, no rotation moves.
  v2f a0, a1; float4 c00, c01, c10, c11;
  loadT(t0, a0, c00, c01);
  for (int t = t0; t < t1; t += 2) {
    loadT(t + 1, a1, c10, c11);                    // t+1 <= t1-1 (tps even)
    process(t, a0, c00, c01);
    int t2 = (t + 2 < t1) ? (t + 2) : (t1 - 1);    // scalar clamp (last iter)
    loadT(t2, a0, c00, c01);
    process(t + 1, a1, c10, c11);
  }

  // Merge the two per-lane sorted lists (lanes n and n+16) of each query.
#pragma unroll
  for (int i = 0; i < KNN; ++i) { sd[wib][lane][i] = bestd[i]; si[wib][lane][i] = besti[i]; }
  __syncthreads();
  if (h == 0) {
    const float* da = sd[wib][n];      const int* ia = si[wib][n];
    const float* db = sd[wib][n + 16]; const int* ib = si[wib][n + 16];
    int i = 0, j = 0;
    size_t obase = ((size_t)q * S + slice) * KNN;
#pragma unroll
    for (int k = 0; k < KNN; ++k) {
      float va = da[i], vb = db[j];
      bool ta = va <= vb;
      part_d[obase + k] = ta ? va : vb;
      part_i[obase + k] = ta ? ia[i] : ib[j];
      i += ta ? 1 : 0; j += ta ? 0 : 1;
    }
  }
}

// ---- Kernel 3: S-way merge of per-slice top-16 lists, emit edges ----------
template <int S>
__global__ void rg_merge(const float* __restrict__ part_d,
                         const int* __restrict__ part_i,
                         int* __restrict__ out_ref, int* __restrict__ out_q,
                         int M) {
  int q = blockIdx.x * blockDim.x + threadIdx.x;
  if (q >= M) return;
  int p[S];
#pragma unroll
  for (int s = 0; s < S; ++s) p[s] = 0;
  const float* bd = part_d + (size_t)q * S * KNN;
  const int*   bi = part_i + (size_t)q * S * KNN;
  for (int k = 0; k < KNN; ++k) {
    float best = SENT; int bs = -1;
#pragma unroll
    for (int s = 0; s < S; ++s) {
      float v = (p[s] < KNN) ? bd[s * KNN + p[s]] : SENT;
      if (v < best) { best = v; bs = s; }
    }
    int rid = -1, qid = -1;
    if (bs >= 0) { rid = bi[bs * KNN + p[bs]]; qid = q; p[bs]++; }
    out_ref[q * KNN + k] = rid;
    out_q[q * KNN + k]   = qid;
  }
}

// ---------------------------------------------------------------------------
extern "C" void kernel_launch(void* const* d_in, const int* in_sizes, int n_in,
                              void* d_out, int out_size, void* d_ws, size_t ws_size,
                              hipStream_t stream) {
  const float4* ref    = (const float4*)d_in[0];
  const float4* query  = (const float4*)d_in[1];
  const float*  radius = (const float*)d_in[2];
  const int N = in_sizes[0] / 4;
  const int M = in_sizes[1] / 4;

  // Pick slice count from available scratch: more slices -> more waves.
  const size_t fixed = (size_t)N * 3 * sizeof(float);
  int S;
  if      (ws_size >= fixed + (size_t)M * 16 * KNN * 8) S = 16;
  else if (ws_size >= fixed + (size_t)M *  8 * KNN * 8) S = 8;
  else if (ws_size >= fixed + (size_t)M *  4 * KNN * 8) S = 4;
  else                                                  S = 2;

  char* ws = (char*)d_ws;
  float* packedA = (float*)ws;                                   // N*2 floats
  float* r2      = (float*)(ws + (size_t)N * 2 * sizeof(float)); // N floats
  float* part_d  = (float*)(ws + fixed);                         // M*S*K
  int*   part_i  = (int*)(ws + fixed + (size_t)M * S * KNN * sizeof(float));
  int* out = (int*)d_out;   // edges[0]=ref_idx, edges[1]=query_idx

  rg_pack<<<(N + 255) / 256, 256, 0, stream>>>(ref, packedA, r2, N);

  const int waves  = (M / 16) * S;
  const int blocks = waves / 8;                  // 8 waves (256 thr) / block
  rg_main<<<blocks, 256, 0, stream>>>(query, radius, packedA, r2,
                                      part_d, part_i, N, S);

  const int mblocks = (M + 255) / 256;
  int* out_q = out + (size_t)M * KNN;
  switch (S) {
    case 16: rg_merge<16><<<mblocks, 256, 0, stream>>>(part_d, part_i, out, out_q, M); break;
    case 8:  rg_merge<8> <<<mblocks, 256, 0, stream>>>(part_d, part_i, out, out_q, M); break;
    case 4:  rg_merge<4> <<<mblocks, 256, 0, stream>>>(part_d, part_i, out, out_q, M); break;
    default: rg_merge<2> <<<mblocks, 256, 0, stream>>>(part_d, part_i, out, out_q, M); break;
  }
}